// MotionPredictor_35579509080389
// MI455X (gfx1250) — compile-verified
//
#include <hip/hip_runtime.h>
#include <math.h>

#define HID 256
#define TT  16
#define NB  4
#define HOR 5

typedef __attribute__((ext_vector_type(2))) float v2f;
typedef __attribute__((ext_vector_type(4))) float fv4;
typedef __attribute__((ext_vector_type(8))) float v8f;

// ---- d_out layout (floats), tuple return order ----
#define OUT_DELTAS 0        // (4,5,4)   = 80
#define OUT_PBBOX  80       // (4,5,4)   = 80
#define OUT_LOC    160      // (4,5,2)   = 40
#define OUT_SCALE  200      // (4,5,2)   = 40
#define OUT_MOTION 240      // (4,16,256)= 16384
#define OUT_CONF   16624    // (4,5)     = 20
#define OUT_BBOX   16644    // (4,16,4)  = 256  -> total 16900

// ---- workspace layout ----
// ints:  wsI[0..255]  (xmin[64], xmax[64], ymin[64], ymax[64])
// X  = wsF + 256          : 64*256  floats (pos2, then layer-0 h sequence)
// XG = wsF + 256 + 16384  : 64*1024 floats (pos1, then xg0, then xg1)

// =======================================================================
// Kernel 0: init min/max accumulators
// =======================================================================
__global__ void ws_init(int* wsI) {
  int i = threadIdx.x;
  if (i < 64) {
    wsI[i]       = 1 << 30;  // xmin
    wsI[64 + i]  = -1;       // xmax
    wsI[128 + i] = 1 << 30;  // ymin
    wsI[192 + i] = -1;       // ymax
  }
}

// =======================================================================
// Kernel 1: bandwidth-bound mask scan. 64 blocks per image, float4 loads,
// non-temporal (predictions = 256MB streamed once; larger than L2).
// =======================================================================
__global__ __launch_bounds__(256)
void bbox_reduce(const float* __restrict__ pred, int* __restrict__ wsI) {
  const int img   = blockIdx.x >> 6;   // 64 images (b*16+t)
  const int chunk = blockIdx.x & 63;   // 64 chunks per image
  const int tid   = threadIdx.x;
  const fv4* p4 = reinterpret_cast<const fv4*>(pred + (size_t)img * 1048576u);

  int xmn = 1 << 30, xmx = -1, ymn = 1 << 30, ymx = -1;
  const int base4 = chunk * 4096;      // 16384 pixels / 4
  #pragma unroll 4
  for (int j = 0; j < 16; ++j) {
    int f4  = base4 + j * 256 + tid;
    fv4 v = __builtin_nontemporal_load(p4 + f4);
    int idx = f4 << 2;
    int y = idx >> 10, x = idx & 1023;
    if (v[0] > 0.5f) { xmn = min(xmn, x);     xmx = max(xmx, x);     ymn = min(ymn, y); ymx = max(ymx, y); }
    if (v[1] > 0.5f) { xmn = min(xmn, x + 1); xmx = max(xmx, x + 1); ymn = min(ymn, y); ymx = max(ymx, y); }
    if (v[2] > 0.5f) { xmn = min(xmn, x + 2); xmx = max(xmx, x + 2); ymn = min(ymn, y); ymx = max(ymx, y); }
    if (v[3] > 0.5f) { xmn = min(xmn, x + 3); xmx = max(xmx, x + 3); ymn = min(ymn, y); ymx = max(ymx, y); }
  }
  // wave32 butterfly reduction
  for (int off = 16; off > 0; off >>= 1) {
    xmn = min(xmn, __shfl_xor(xmn, off, 32));
    xmx = max(xmx, __shfl_xor(xmx, off, 32));
    ymn = min(ymn, __shfl_xor(ymn, off, 32));
    ymx = max(ymx, __shfl_xor(ymx, off, 32));
  }
  __shared__ int s[4][8];
  int wv = tid >> 5, ln = tid & 31;
  if (ln == 0) { s[0][wv] = xmn; s[1][wv] = xmx; s[2][wv] = ymn; s[3][wv] = ymx; }
  __syncthreads();
  if (tid == 0) {
    for (int w = 1; w < 8; ++w) {
      xmn = min(xmn, s[0][w]); xmx = max(xmx, s[1][w]);
      ymn = min(ymn, s[2][w]); ymx = max(ymx, s[3][w]);
    }
    atomicMin(&wsI[img],        xmn);
    atomicMax(&wsI[64  + img],  xmx);
    atomicMin(&wsI[128 + img],  ymn);
    atomicMax(&wsI[192 + img],  ymx);
  }
}

// =======================================================================
// WMMA f32 16x16x4 tile: C(16x16) += A(16x4-slices of K) * B(Kx16)
// A: row-major, stride lda (global).  W: row-major KxN, stride ldw (global).
// ISA layouts: a[v]=A[l%16][k+v+2*(l>=16)], b[v]=B[k+v+2*(l>=16)][l%16]
// =======================================================================
__device__ __forceinline__ v8f tile_gemm(const float* __restrict__ Ab, int lda,
                                         const float* __restrict__ Wb, int ldw,
                                         int K, int lane) {
  const int lh = lane >> 4;
  const int ll = lane & 15;
  const float* arow = Ab + ll * lda;
  v8f acc = {};
  for (int k = 0; k < K; k += 4) {
    int k0 = k + 2 * lh;
    float2 av = *reinterpret_cast<const float2*>(arow + k0);
    v2f a; a[0] = av.x; a[1] = av.y;
    v2f b; b[0] = Wb[k0 * ldw + ll]; b[1] = Wb[(k0 + 1) * ldw + ll];
    acc = __builtin_amdgcn_wmma_f32_16x16x4_f32(false, a, false, b, (short)0, acc,
                                                false, false);
  }
  return acc;
}

__device__ __forceinline__ float sigmoidf_(float x) { return 1.f / (1.f + expf(-x)); }

// =======================================================================
// Kernel 2: whole network in one workgroup (1024 threads = 32 wave32).
// =======================================================================
__global__ __launch_bounds__(1024)
void mp_main(const float* __restrict__ pe_w1, const float* __restrict__ pe_b1,
             const float* __restrict__ pe_w2, const float* __restrict__ pe_b2,
             const float* __restrict__ wih0,  const float* __restrict__ whh0,
             const float* __restrict__ bih0,  const float* __restrict__ bhh0,
             const float* __restrict__ wih1,  const float* __restrict__ whh1,
             const float* __restrict__ bih1,  const float* __restrict__ bhh1,
             const float* __restrict__ mp_w1, const float* __restrict__ mp_b1,
             const float* __restrict__ mp_w2, const float* __restrict__ mp_b2,
             const float* __restrict__ mp_w3, const float* __restrict__ mp_b3,
             const float* __restrict__ cf_w1, const float* __restrict__ cf_b1,
             const float* __restrict__ cf_w2, const float* __restrict__ cf_b2,
             float* __restrict__ out, const int* __restrict__ wsI,
             float* __restrict__ X, float* __restrict__ XG) {
  __shared__ float hmat[16 * HID];   // h, rows 0..3 live, 4..15 zero (M padded to 16)
  __shared__ float cbuf[NB * HID];   // c
  __shared__ float gbuf[NB * 1024];  // gates / head scratch
  __shared__ float bb[64 * 4];       // bboxes
  __shared__ float dbuf[NB * HOR * 4];

  const int tid  = threadIdx.x;
  const int lane = tid & 31;
  const int wave = tid >> 5;
  const int lh   = lane >> 4;
  const int ll   = lane & 15;

  // ---- (a) bboxes from reduction results ----
  if (tid < 64) {
    int xmn = wsI[tid], xmx = wsI[64 + tid], ymn = wsI[128 + tid], ymx = wsI[192 + tid];
    float b0, b1, b2, b3;
    if (xmx >= 0) {
      b0 = (float)(xmn + xmx) * 0.5f * (1.f / 1024.f);
      b1 = (float)(ymn + ymx) * 0.5f * (1.f / 1024.f);
      b2 = (float)(xmx - xmn) * (1.f / 1024.f);
      b3 = (float)(ymx - ymn) * (1.f / 1024.f);
    } else { b0 = 0.5f; b1 = 0.5f; b2 = 0.1f; b3 = 0.1f; }
    bb[tid * 4 + 0] = b0; bb[tid * 4 + 1] = b1; bb[tid * 4 + 2] = b2; bb[tid * 4 + 3] = b3;
    out[OUT_BBOX + tid * 4 + 0] = b0; out[OUT_BBOX + tid * 4 + 1] = b1;
    out[OUT_BBOX + tid * 4 + 2] = b2; out[OUT_BBOX + tid * 4 + 3] = b3;
  }
  __syncthreads();

  // ---- (b) pos1 = relu(bb @ pe_w1 + pe_b1) -> XG[64][256] ----
  for (int idx = tid; idx < 64 * HID; idx += 1024) {
    int row = idx >> 8, n = idx & 255;
    float s = pe_b1[n]
            + bb[row * 4 + 0] * pe_w1[n]
            + bb[row * 4 + 1] * pe_w1[256 + n]
            + bb[row * 4 + 2] * pe_w1[512 + n]
            + bb[row * 4 + 3] * pe_w1[768 + n];
    XG[idx] = fmaxf(s, 0.f);
  }
  __threadfence(); __syncthreads();

  // ---- (c) pos2 = relu(pos1 @ pe_w2 + pe_b2) -> X[64][256], WMMA ----
  for (int ti = 0; ti < 2; ++ti) {
    int tile = wave * 2 + ti;
    int mt = tile >> 4, nt = tile & 15;
    v8f acc = tile_gemm(XG + mt * 16 * HID, HID, pe_w2 + nt * 16, HID, HID, lane);
    int col = nt * 16 + ll;
    float bias = pe_b2[col];
    for (int r = 0; r < 8; ++r)
      X[(mt * 16 + r + 8 * lh) * HID + col] = fmaxf(acc[r] + bias, 0.f);
  }
  __threadfence(); __syncthreads();

  // ---- (d) xg0 = pos2 @ wih0 + bih0 -> XG[64][1024], WMMA ----
  for (int ti = 0; ti < 8; ++ti) {
    int tile = wave * 8 + ti;
    int mt = tile >> 6, nt = tile & 63;
    v8f acc = tile_gemm(X + mt * 16 * HID, HID, wih0 + nt * 16, 1024, HID, lane);
    int col = nt * 16 + ll;
    float bias = bih0[col];
    for (int r = 0; r < 8; ++r)
      XG[(mt * 16 + r + 8 * lh) * 1024 + col] = acc[r] + bias;
  }
  for (int i = tid; i < 16 * HID; i += 1024) hmat[i] = 0.f;
  for (int i = tid; i < NB * HID; i += 1024) cbuf[i] = 0.f;
  __threadfence(); __syncthreads();

  // ---- (e) two LSTM layers, serial over T ----
  for (int layer = 0; layer < 2; ++layer) {
    const float* whh = layer ? whh1 : whh0;
    const float* bhh = layer ? bhh1 : bhh0;
    for (int t = 0; t < TT; ++t) {
      // h @ whh : M=16(pad), N=1024 -> 64 tiles, 2 per wave; A from LDS
      for (int ti = 0; ti < 2; ++ti) {
        int nt = wave * 2 + ti;
        const float* wb = whh + nt * 16;
        v8f acc = {};
        for (int k = 0; k < HID; k += 4) {
          int k0 = k + 2 * lh;
          v2f a; a[0] = hmat[ll * HID + k0]; a[1] = hmat[ll * HID + k0 + 1];
          v2f b; b[0] = wb[k0 * 1024 + ll];  b[1] = wb[(k0 + 1) * 1024 + ll];
          acc = __builtin_amdgcn_wmma_f32_16x16x4_f32(false, a, false, b, (short)0,
                                                      acc, false, false);
        }
        if (lane < 16) {  // only batch rows 0..3 are real
          int col = nt * 16 + lane;
          for (int r = 0; r < 4; ++r) gbuf[r * 1024 + col] = acc[r];
        }
      }
      __syncthreads();
      // gates: g = xg[t] + h@whh + bhh ; one (b,n) per thread
      {
        int b = tid >> 8, n = tid & 255;
        int row = b * TT + t;
        const float* xgr = XG + row * 1024;
        float gi = gbuf[b * 1024 + n]        + xgr[n]        + bhh[n];
        float gf = gbuf[b * 1024 + 256 + n]  + xgr[256 + n]  + bhh[256 + n];
        float gg = gbuf[b * 1024 + 512 + n]  + xgr[512 + n]  + bhh[512 + n];
        float go = gbuf[b * 1024 + 768 + n]  + xgr[768 + n]  + bhh[768 + n];
        gi = sigmoidf_(gi); gf = sigmoidf_(gf); go = sigmoidf_(go);
        gg = tanhf(gg);
        float c = gf * cbuf[b * HID + n] + gi * gg;
        cbuf[b * HID + n] = c;
        float h = go * tanhf(c);
        hmat[b * HID + n] = h;
        if (layer == 0) X[row * HID + n] = h;                    // h1 sequence
        else            out[OUT_MOTION + row * HID + n] = h;     // motion_features
      }
      __syncthreads();
    }
    if (layer == 0) {
      __threadfence(); __syncthreads();
      // xg1 = h1seq @ wih1 + bih1 -> XG, WMMA
      for (int ti = 0; ti < 8; ++ti) {
        int tile = wave * 8 + ti;
        int mt = tile >> 6, nt = tile & 63;
        v8f acc = tile_gemm(X + mt * 16 * HID, HID, wih1 + nt * 16, 1024, HID, lane);
        int col = nt * 16 + ll;
        float bias = bih1[col];
        for (int r = 0; r < 8; ++r)
          XG[(mt * 16 + r + 8 * lh) * 1024 + col] = acc[r] + bias;
      }
      for (int i = tid; i < 16 * HID; i += 1024) hmat[i] = 0.f;
      for (int i = tid; i < NB * HID; i += 1024) cbuf[i] = 0.f;
      __threadfence(); __syncthreads();
    }
  }
  __syncthreads();   // hmat rows 0..3 = last_hidden

  // ---- (f) heads ----
  // mp1: relu(h @ mp_w1 + b1) -> gbuf[0..1023]
  {
    int b = tid >> 8, n = tid & 255;
    float s = mp_b1[n];
    for (int k = 0; k < HID; ++k) s += hmat[b * HID + k] * mp_w1[k * HID + n];
    gbuf[b * 256 + n] = fmaxf(s, 0.f);
  }
  __syncthreads();
  // mp2 -> gbuf[1024..1535]; cf1 -> gbuf[1536..2047]
  if (tid < 512) {
    int b = tid >> 7, n = tid & 127;
    float s = mp_b2[n];
    for (int k = 0; k < HID; ++k) s += gbuf[b * 256 + k] * mp_w2[k * 128 + n];
    gbuf[1024 + b * 128 + n] = fmaxf(s, 0.f);
  } else {
    int u = tid - 512; int b = u >> 7, n = u & 127;
    float s = cf_b1[n];
    for (int k = 0; k < HID; ++k) s += hmat[b * HID + k] * cf_w1[k * 128 + n];
    gbuf[1536 + b * 128 + n] = fmaxf(s, 0.f);
  }
  __syncthreads();
  // mp3 -> deltas; cf2 -> confidence
  if (tid < 80) {
    int b = tid / 20, j = tid % 20;
    float s = mp_b3[j];
    for (int k = 0; k < 128; ++k) s += gbuf[1024 + b * 128 + k] * mp_w3[k * 20 + j];
    out[OUT_DELTAS + b * 20 + j] = s;
    dbuf[b * 20 + j] = s;
  } else if (tid >= 128 && tid < 148) {
    int u = tid - 128; int b = u / 5, j = u % 5;
    float s = cf_b2[j];
    for (int k = 0; k < 128; ++k) s += gbuf[1536 + b * 128 + k] * cf_w2[k * 5 + j];
    out[OUT_CONF + b * 5 + j] = sigmoidf_(s);
  }
  __syncthreads();
  // cumsum: pred_bboxes = last_bbox + cumsum(deltas)
  if (tid < 16) {
    int b = tid >> 2, j = tid & 3;
    float last = bb[(b * 16 + 15) * 4 + j];
    for (int k = 0; k < HOR; ++k) {
      last += dbuf[b * 20 + k * 4 + j];
      out[OUT_PBBOX + b * 20 + k * 4 + j] = last;
      if (j < 2) out[OUT_LOC   + b * 10 + k * 2 + j]       = last;
      else       out[OUT_SCALE + b * 10 + k * 2 + (j - 2)] = last;
    }
  }
}

// =======================================================================
extern "C" void kernel_launch(void* const* d_in, const int* in_sizes, int n_in,
                              void* d_out, int out_size, void* d_ws, size_t ws_size,
                              hipStream_t stream) {
  // inputs: 0 features(unused) 1 predictions 2..23 weights/biases
  const float* pred  = (const float*)d_in[1];
  const float* pe_w1 = (const float*)d_in[2],  *pe_b1 = (const float*)d_in[3];
  const float* pe_w2 = (const float*)d_in[4],  *pe_b2 = (const float*)d_in[5];
  const float* wih0  = (const float*)d_in[6],  *whh0  = (const float*)d_in[7];
  const float* bih0  = (const float*)d_in[8],  *bhh0  = (const float*)d_in[9];
  const float* wih1  = (const float*)d_in[10], *whh1  = (const float*)d_in[11];
  const float* bih1  = (const float*)d_in[12], *bhh1  = (const float*)d_in[13];
  const float* mp_w1 = (const float*)d_in[14], *mp_b1 = (const float*)d_in[15];
  const float* mp_w2 = (const float*)d_in[16], *mp_b2 = (const float*)d_in[17];
  const float* mp_w3 = (const float*)d_in[18], *mp_b3 = (const float*)d_in[19];
  const float* cf_w1 = (const float*)d_in[20], *cf_b1 = (const float*)d_in[21];
  const float* cf_w2 = (const float*)d_in[22], *cf_b2 = (const float*)d_in[23];

  float* out = (float*)d_out;
  int*   wsI = (int*)d_ws;
  float* wsF = (float*)d_ws;
  float* X   = wsF + 256;             // 64*256
  float* XG  = wsF + 256 + 64 * 256;  // 64*1024

  ws_init<<<1, 64, 0, stream>>>(wsI);
  bbox_reduce<<<4096, 256, 0, stream>>>(pred, wsI);
  mp_main<<<1, 1024, 0, stream>>>(pe_w1, pe_b1, pe_w2, pe_b2,
                                  wih0, whh0, bih0, bhh0,
                                  wih1, whh1, bih1, bhh1,
                                  mp_w1, mp_b1, mp_w2, mp_b2, mp_w3, mp_b3,
                                  cf_w1, cf_b1, cf_w2, cf_b2,
                                  out, wsI, X, XG);
}